// HeatmapOffsetmapLoss_75625784148260
// MI455X (gfx1250) — compile-verified
//
#include <hip/hip_runtime.h>

typedef __attribute__((ext_vector_type(2))) float v2f;
typedef __attribute__((ext_vector_type(8))) float v8f;

#define Hh      512
#define Ww      512
#define Ll      19
#define Bb      8
#define PAIRS   (Bb * Ll)      // 152
#define CHUNKS  16
#define THREADS 256
#define ITERS   16             // 16 iters * 256 thr * 4 floats = 16384 px = 262144/16

__device__ __forceinline__ int clamp_round511(float v) {
    // jnp.round is round-half-to-even; rintf uses RNE (hardware default) -> matches
    float r = rintf(v * 511.0f);
    r = fminf(fmaxf(r, 0.0f), 511.0f);
    return (int)r;
}

// Pass 1: each block reduces one (pair, chunk) slab of 16384 pixels across 3 planes.
__global__ void __launch_bounds__(THREADS)
heatmap_loss_partial(const float* __restrict__ fm,
                     const float* __restrict__ lm,
                     float4* __restrict__ partials) {
    const int blk   = blockIdx.x;
    const int pair  = blk / CHUNKS;
    const int chunk = blk - pair * CHUNKS;
    const int b     = pair / Ll;
    const int l     = pair - b * Ll;

    const int X = clamp_round511(lm[pair * 2 + 0]);
    const int Y = clamp_round511(lm[pair * 2 + 1]);

    const size_t plane  = (size_t)Hh * Ww;
    const size_t baseL  = ((size_t)b * 57 + l) * plane;
    const float4* pl = (const float4*)(fm + baseL);
    const float4* px = (const float4*)(fm + baseL + (size_t)Ll * plane);
    const float4* py = (const float4*)(fm + baseL + (size_t)(2 * Ll) * plane);

    const int   base  = chunk * (THREADS * ITERS);   // float4 index into plane
    const float inv41 = 1.0f / 41.0f;

    float bce = 0.0f, sx = 0.0f, sy = 0.0f;
    int   cnt = 0;

#pragma unroll 2
    for (int it = 0; it < ITERS; ++it) {
        const int q = base + it * THREADS + threadIdx.x;
        if (it + 2 < ITERS) {                       // gfx1250 global_prefetch_b8
            __builtin_prefetch(&pl[q + 2 * THREADS], 0, 0);
            __builtin_prefetch(&px[q + 2 * THREADS], 0, 0);
            __builtin_prefetch(&py[q + 2 * THREADS], 0, 0);
        }
        const float4 lv = pl[q];
        const float4 xv = px[q];
        const float4 yv = py[q];

        const int p   = q << 2;                     // pixel index
        const int i   = p >> 9;                     // row
        const int j0  = p & (Ww - 1);               // col of element 0
        const int dx  = i - X;
        const int dx2 = dx * dx;
        const float offx = (float)(X - i) * inv41;  // (X - i)/41

        const float lg[4] = {lv.x, lv.y, lv.z, lv.w};
        const float fx[4] = {xv.x, xv.y, xv.z, xv.w};
        const float fy[4] = {yv.x, yv.y, yv.z, yv.w};

#pragma unroll
        for (int e = 0; e < 4; ++e) {
            const int  dy = (j0 + e) - Y;
            const bool m  = (dx2 + dy * dy) <= 41 * 41;   // heat > 0 (inclusive)
            const float g = lg[e];
            // BCE-with-logits: max(g,0) - g*heat + log(1 + exp(-|g|))
            // native v_exp_f32 / v_log_f32; t in [~6.7e-3, 1] for N(0,1) inputs,
            // so log(1+t) is numerically safe (log1p only matters for t -> 0).
            const float t = __expf(-fabsf(g));
            bce += fmaxf(g, 0.0f) - (m ? g : 0.0f) + __logf(1.0f + t);
            if (m) {
                cnt += 1;
                sx += fabsf(fx[e] - offx);
                // offy = (Y-j)/41  =>  fy - offy = fy + dy/41
                sy += fabsf(fy[e] + (float)dy * inv41);
            }
        }
    }

    __shared__ float4 red[THREADS];
    red[threadIdx.x] = make_float4(bce, sx, sy, (float)cnt);
    __syncthreads();
#pragma unroll
    for (int s = THREADS / 2; s > 0; s >>= 1) {
        if (threadIdx.x < (unsigned)s) {
            float4 a = red[threadIdx.x];
            float4 c = red[threadIdx.x + s];
            red[threadIdx.x] = make_float4(a.x + c.x, a.y + c.y, a.z + c.z, a.w + c.w);
        }
        __syncthreads();
    }
    if (threadIdx.x == 0) partials[blk] = red[0];
}

// Pass 2: combine per-pair partials, then reduce 256 per-pair losses with
// V_WMMA_F32_16X16X4_F32 (D = A x ones + C, exact in f32).
__global__ void __launch_bounds__(256)
heatmap_loss_final(const float* __restrict__ lm,
                   const float4* __restrict__ partials,
                   float* __restrict__ out) {
    __shared__ float vals[256];
    const int t = threadIdx.x;

    float lossv = 0.0f;
    if (t < PAIRS) {
        const int X = clamp_round511(lm[t * 2 + 0]);
        const int Y = clamp_round511(lm[t * 2 + 1]);
        float bce = 0.0f, sx = 0.0f, sy = 0.0f, cnt = 0.0f;
        for (int c = 0; c < CHUNKS; ++c) {
            float4 v = partials[t * CHUNKS + c];
            bce += v.x; sx += v.y; sy += v.z; cnt += v.w;
        }
        const float cc    = fmaxf(cnt, 1.0f);
        const float valid = (X > 0 && Y > 0) ? 1.0f : 0.0f;
        lossv = valid * (2.0f * bce * (1.0f / (float)(Hh * Ww)) + (sx + sy) / cc);
    }
    vals[t] = lossv;
    __syncthreads();

    if (t < 32) {   // wave 0: EXEC all-ones (WMMA requirement)
        v8f acc = 0.0f;
        v2f ones; ones.x = 1.0f; ones.y = 1.0f;   // B = ones -> layout irrelevant
        const int m  = t & 15;
        const int kb = (t >> 4) << 1;             // A 16x4 f32: lanes16-31 hold K=2,3
#pragma unroll
        for (int c = 0; c < 4; ++c) {             // 4 x 64 values = 256
            v2f a;
            a.x = vals[c * 64 + m * 4 + kb];
            a.y = vals[c * 64 + m * 4 + kb + 1];
            acc = __builtin_amdgcn_wmma_f32_16x16x4_f32(
                false, a, false, ones, (short)0, acc, false, false);
        }
        // D[m][n] = row-sum of A for all n; column n=0 lives in lane0 (M=0..7)
        // and lane16 (M=8..15) across the 8 acc VGPRs.
        float s = acc[0] + acc[1] + acc[2] + acc[3] + acc[4] + acc[5] + acc[6] + acc[7];
        float s0  = __shfl(s, 0, 32);
        float s16 = __shfl(s, 16, 32);
        if (t == 0) out[0] = (s0 + s16) * (1.0f / (float)PAIRS);
    }
}

extern "C" void kernel_launch(void* const* d_in, const int* in_sizes, int n_in,
                              void* d_out, int out_size, void* d_ws, size_t ws_size,
                              hipStream_t stream) {
    const float* fm = (const float*)d_in[0];   // featureMaps [8,57,512,512] f32
    const float* lm = (const float*)d_in[1];   // landmarks   [8,19,2]       f32
    float*       out = (float*)d_out;          // scalar f32
    float4*      partials = (float4*)d_ws;     // needs 152*16*16 = 38912 bytes

    heatmap_loss_partial<<<PAIRS * CHUNKS, THREADS, 0, stream>>>(fm, lm, partials);
    heatmap_loss_final<<<1, 256, 0, stream>>>(lm, partials, out);
}